// GCN_56762287784225
// MI455X (gfx1250) — compile-verified
//
#include <hip/hip_runtime.h>
#include <hip/hip_bf16.h>

// y = (w0*I + w1*L + w2*L^2) x,  L = D - A  (graph Laplacian)
// N=100000 nodes, C=32 channels, E edges (~1.6M).
//
// Strategy (L2-resident, atomic-bound):
//   K0: zero deg[N], agg[N*C]
//   K1: per-edge scatter: agg[dst*32+c] += x[src*32+c]; deg[dst] += 1 (lane 0)
//   K2: combine (WMMA): t1 = diag(deg)@X - agg  via V_WMMA_F32_16X16X4_F32 chain,
//       y = w0*x + w1*t1 -> d_out, t1 -> ws, and agg zeroed in place
//   K3: per-edge scatter on t1
//   K4: combine (WMMA): t2 = diag(deg)@t1 - agg, y = y + w2*t2 -> d_out

typedef __attribute__((ext_vector_type(2))) float v2f;
typedef __attribute__((ext_vector_type(8))) float v8f;

#define GCN_C 32

__global__ __launch_bounds__(256) void gcn_zero2(float* __restrict__ a, long na,
                                                 float* __restrict__ b, long nb) {
  long i = (long)blockIdx.x * blockDim.x + threadIdx.x;
  if (i < na) {
    a[i] = 0.0f;
  } else {
    long j = i - na;
    if (j < nb) b[j] = 0.0f;
  }
}

// One edge per wave32; lane c handles channel c. Coalesced 128B gather +
// 128B contiguous atomic burst per edge. Native f32 atomics (no CAS loop).
__global__ __launch_bounds__(256) void gcn_scatter(const float* __restrict__ t,
                                                   const int* __restrict__ esrc,
                                                   const int* __restrict__ edst,
                                                   float* __restrict__ agg,
                                                   float* __restrict__ deg,  // may be null
                                                   int E) {
  int tid = blockIdx.x * blockDim.x + threadIdx.x;
  int e = tid >> 5;
  int c = tid & 31;
  if (e >= E) return;
  int s = esrc[e];
  int d = edst[e];
  float v = t[s * GCN_C + c];
  unsafeAtomicAdd(&agg[d * GCN_C + c], v);
  if (deg != nullptr && c == 0) unsafeAtomicAdd(&deg[d], 1.0f);
}

// One wave per 16x16 tile (16 nodes x 16 channels; C=32 -> 2 ch-tiles/node-tile).
// t_new = diag(deg) @ T + (-agg)  via 4 chained V_WMMA_F32_16X16X4_F32 (K=16).
// Exact in f32: every D element is a single real product plus zeros.
// EXEC is all-ones for active waves (guard is wave-uniform; N % 16 == 0).
__global__ __launch_bounds__(256) void gcn_combine(const float* __restrict__ t_in,
                                                   const float* __restrict__ x_for_y, // used iff y_prev==null
                                                   float* __restrict__ agg,           // read; zeroed if zero_agg
                                                   const float* __restrict__ deg,
                                                   const float* __restrict__ wts,     // 3 weights on device
                                                   float* __restrict__ y_out,
                                                   const float* __restrict__ y_prev,  // null on round 1
                                                   float* __restrict__ t_out,         // null on last round
                                                   int n_nodes, int wa_idx, int wb_idx,
                                                   int zero_agg) {
  int wave = (blockIdx.x * blockDim.x + threadIdx.x) >> 5;
  int lane = threadIdx.x & 31;
  int node_base = (wave >> 1) * 16;
  int ch_base = (wave & 1) * 16;
  if (node_base >= n_nodes) return;  // wave-uniform

  int g = lane >> 4;    // half-wave group
  int mn = lane & 15;   // A-row index m / B,C,D column index n

  float w_b = wts[wb_idx];
  float dg = deg[node_base + mn];  // diagonal value for A row m=mn

  // C operand = -agg tile in C/D layout: VGPR r <-> m = r + 8g, lane <-> n = mn.
  v8f acc;
#pragma unroll
  for (int r = 0; r < 8; ++r) {
    int idx = (node_base + r + 8 * g) * GCN_C + ch_base + mn;
    acc[r] = -agg[idx];
    if (zero_agg) agg[idx] = 0.0f;  // each element touched by exactly one lane
  }

  // K=16 as 4 slices of V_WMMA_F32_16X16X4_F32.
  // A 16x4 layout: lane m=mn, vgpr v holds K = 2g + v (within slice).
  // B 4x16 layout (mirror): lane n=mn, vgpr v holds row K = 2g + v.
#pragma unroll
  for (int j = 0; j < 4; ++j) {
    int k0 = 4 * j + 2 * g;  // global K of vgpr0 for this lane
    v2f a, b;
    a.x = (k0 == mn) ? dg : 0.0f;      // diag(deg)
    a.y = (k0 + 1 == mn) ? dg : 0.0f;
    b.x = t_in[(node_base + k0) * GCN_C + ch_base + mn];
    b.y = t_in[(node_base + k0 + 1) * GCN_C + ch_base + mn];
    acc = __builtin_amdgcn_wmma_f32_16x16x4_f32(
        /*neg_a=*/false, a, /*neg_b=*/false, b,
        /*c_mod=*/(short)0, acc, /*reuse_a=*/false, /*reuse_b=*/false);
  }

  // acc = t_new tile in C/D layout. Emit t_out and y.
  float w_a = (y_prev == nullptr) ? wts[wa_idx] : 0.0f;
#pragma unroll
  for (int r = 0; r < 8; ++r) {
    int idx = (node_base + r + 8 * g) * GCN_C + ch_base + mn;
    float tn = acc[r];
    if (t_out != nullptr) t_out[idx] = tn;
    float ybase = (y_prev != nullptr) ? y_prev[idx] : (w_a * x_for_y[idx]);
    y_out[idx] = ybase + w_b * tn;
  }
}

extern "C" void kernel_launch(void* const* d_in, const int* in_sizes, int n_in,
                              void* d_out, int out_size, void* d_ws, size_t ws_size,
                              hipStream_t stream) {
  const float* x = (const float*)d_in[0];
  const float* w = (const float*)d_in[1];      // 3 weights
  const int* esrc = (const int*)d_in[2];
  const int* edst = (const int*)d_in[3];
  const int E = in_sizes[2];
  const int C = GCN_C;
  const int N = out_size / C;                  // 100000
  float* out = (float*)d_out;

  float* deg = (float*)d_ws;                   // N floats
  float* agg = deg + N;                        // N*C floats
  float* t1 = agg + (size_t)N * C;             // N*C floats

  // K0: zero deg + agg
  {
    long tot = (long)N + (long)N * C;
    int blocks = (int)((tot + 255) / 256);
    gcn_zero2<<<blocks, 256, 0, stream>>>(deg, (long)N, agg, (long)N * C);
  }

  const long sthreads = (long)E * 32;
  const int sblocks = (int)((sthreads + 255) / 256);
  const int tiles = (N / 16) * 2;
  const int cblocks = (tiles + 7) / 8;  // 8 waves per 256-thread block

  // K1: round-1 scatter (+deg)
  gcn_scatter<<<sblocks, 256, 0, stream>>>(x, esrc, edst, agg, deg, E);

  // K2: t1 = deg*x - agg ; y = w0*x + w1*t1 ; agg := 0
  gcn_combine<<<cblocks, 256, 0, stream>>>(x, x, agg, deg, w, out,
                                           /*y_prev=*/nullptr, /*t_out=*/t1,
                                           N, /*wa=*/0, /*wb=*/1, /*zero_agg=*/1);

  // K3: round-2 scatter on t1
  gcn_scatter<<<sblocks, 256, 0, stream>>>(t1, esrc, edst, agg, /*deg=*/nullptr, E);

  // K4: t2 = deg*t1 - agg ; y = y + w2*t2
  gcn_combine<<<cblocks, 256, 0, stream>>>(t1, /*x_for_y=*/nullptr, agg, deg, w, out,
                                           /*y_prev=*/out, /*t_out=*/nullptr,
                                           N, /*wa=*/0, /*wb=*/2, /*zero_agg=*/0);
}